// HeadTailBoundaryPredictor_13030930776500
// MI455X (gfx1250) — compile-verified
//
#include <hip/hip_runtime.h>
#include <hip/hip_bf16.h>

typedef __attribute__((ext_vector_type(2))) float v2f;
typedef __attribute__((ext_vector_type(8))) float v8f;

static constexpr int Bq = 8, Sq = 512, Eq = 16, Hq = 768;

#define WMMA_F32(acc, a, b) \
    __builtin_amdgcn_wmma_f32_16x16x4_f32(false, (a), false, (b), (short)0, (acc), false, false)

#if defined(__has_builtin)
#if __has_builtin(__builtin_amdgcn_sched_group_barrier)
// Per-iteration schedule: batch the 5 vector loads, then the 4 WMMAs.
// 0x20 = VMEM read, 0x08 = MFMA/WMMA.
#define SCHED_PIPELINE()                                    \
    do {                                                    \
        __builtin_amdgcn_sched_group_barrier(0x020, 5, 0);  \
        __builtin_amdgcn_sched_group_barrier(0x008, 4, 0);  \
    } while (0)
#endif
#endif
#ifndef SCHED_PIPELINE
#define SCHED_PIPELINE() do {} while (0)
#endif

// Generic GEMM: out[m][n] = sum_k A[m][k] * W[n][k] + bias[n]
// A: M x K row-major (K contiguous), W: N x K row-major (K contiguous).
// Each wave computes a 16x64 output tile via V_WMMA_F32_16X16X4_F32 with a
// distance-2 software pipeline (3 fragment sets live, ~10 loads in flight)
// and sched_group_barrier hints so the scheduler keeps [loads][wmma] batches
// instead of load->wait(0)->wmma lockstep.
__global__ __launch_bounds__(256) void gemm_bias_wmma_f32(
    const float* __restrict__ A, const float* __restrict__ W,
    const float* __restrict__ bias, float* __restrict__ out,
    int M, int N, int K, int nTilesN, int totalTiles)
{
    const int wave = threadIdx.x >> 5;
    const int lane = threadIdx.x & 31;
    const int tile = blockIdx.x * 8 + wave;
    if (tile >= totalTiles) return;          // whole-wave exit: EXEC all-1s for WMMA

    const int mtile = tile / nTilesN;
    const int ntile = tile % nTilesN;

    const int row   = mtile * 16 + (lane & 15);
    const int nbase = ntile * 64 + (lane & 15);
    const int koff  = (lane >> 4) << 1;      // 0 for lanes 0-15, 2 for lanes 16-31

    const float* ap  = A + (size_t)row * K + koff;
    const float* wp0 = W + (size_t)(nbase +  0) * K + koff;
    const float* wp1 = W + (size_t)(nbase + 16) * K + koff;
    const float* wp2 = W + (size_t)(nbase + 32) * K + koff;
    const float* wp3 = W + (size_t)(nbase + 48) * K + koff;

    v8f acc0 = {}, acc1 = {}, acc2 = {}, acc3 = {};

    // Pipeline prologue: sets for k = 0 and k = 4.
    v2f a0  = *(const v2f*)(ap);
    v2f b00 = *(const v2f*)(wp0);
    v2f b01 = *(const v2f*)(wp1);
    v2f b02 = *(const v2f*)(wp2);
    v2f b03 = *(const v2f*)(wp3);

    v2f a1  = *(const v2f*)(ap  + 4);
    v2f b10 = *(const v2f*)(wp0 + 4);
    v2f b11 = *(const v2f*)(wp1 + 4);
    v2f b12 = *(const v2f*)(wp2 + 4);
    v2f b13 = *(const v2f*)(wp3 + 4);

#pragma unroll 4
    for (int k = 8; k < K; k += 4) {
        // Issue loads two steps ahead of consumption.
        v2f a2  = *(const v2f*)(ap  + k);
        v2f b20 = *(const v2f*)(wp0 + k);
        v2f b21 = *(const v2f*)(wp1 + k);
        v2f b22 = *(const v2f*)(wp2 + k);
        v2f b23 = *(const v2f*)(wp3 + k);

        acc0 = WMMA_F32(acc0, a0, b00);
        acc1 = WMMA_F32(acc1, a0, b01);
        acc2 = WMMA_F32(acc2, a0, b02);
        acc3 = WMMA_F32(acc3, a0, b03);

        a0 = a1;  b00 = b10; b01 = b11; b02 = b12; b03 = b13;
        a1 = a2;  b10 = b20; b11 = b21; b12 = b22; b13 = b23;

        SCHED_PIPELINE();
    }

    // Pipeline epilogue: last two K-steps.
    acc0 = WMMA_F32(acc0, a0, b00);
    acc1 = WMMA_F32(acc1, a0, b01);
    acc2 = WMMA_F32(acc2, a0, b02);
    acc3 = WMMA_F32(acc3, a0, b03);

    acc0 = WMMA_F32(acc0, a1, b10);
    acc1 = WMMA_F32(acc1, a1, b11);
    acc2 = WMMA_F32(acc2, a1, b12);
    acc3 = WMMA_F32(acc3, a1, b13);

    // D layout: VGPR r holds M = r (lanes 0-15) / r+8 (lanes 16-31), N = lane&15.
    const int m0   = mtile * 16 + ((lane >> 4) << 3);
    const int ncol = ntile * 64 + (lane & 15);
    const float bv0 = bias[ncol];
    const float bv1 = bias[ncol + 16];
    const float bv2 = bias[ncol + 32];
    const float bv3 = bias[ncol + 48];

#pragma unroll
    for (int r = 0; r < 8; ++r) {
        float* o = out + (size_t)(m0 + r) * N;
        o[ncol]      = acc0[r] + bv0;
        o[ncol + 16] = acc1[r] + bv1;
        o[ncol + 32] = acc2[r] + bv2;
        o[ncol + 48] = acc3[r] + bv3;
    }
}

// One block per (b,s). Stage t[b,s,:] and wb in LDS; one wave per entity,
// 32-lane shuffle reduction; then mask + sigmoid.
__global__ __launch_bounds__(256) void score_kernel(
    const float* __restrict__ t, const float* __restrict__ e,
    const unsigned char* __restrict__ mask, const float* __restrict__ wb,
    float* __restrict__ cls_out, float* __restrict__ p_out)
{
    __shared__ float t_lds[Hq];
    __shared__ float w_lds[Hq];

    const int bs = blockIdx.x;           // b*Sq + s
    const int b  = bs / Sq;
    const int s  = bs % Sq;

    for (int i = threadIdx.x; i < Hq; i += 256) {
        t_lds[i] = t[(size_t)bs * Hq + i];
        w_lds[i] = wb[i];
    }
    __syncthreads();

    const int wave = threadIdx.x >> 5;
    const int lane = threadIdx.x & 31;
    const bool valid = mask[bs] != 0;

    for (int ei = wave; ei < Eq; ei += 8) {
        const float* erow = e + (size_t)(b * Eq + ei) * Hq;
        float sum = 0.0f;
#pragma unroll 4
        for (int h = lane; h < Hq; h += 32) {
            float v = t_lds[h] + erow[h];
            v = v > 0.0f ? v : 0.0f;
            sum = fmaf(v, w_lds[h], sum);
        }
#pragma unroll
        for (int off = 16; off > 0; off >>= 1)
            sum += __shfl_xor(sum, off, 32);

        if (lane == 0) {
            const float c = valid ? sum : -10000.0f;
            const size_t oidx = (size_t)(b * Eq + ei) * Sq + s;
            cls_out[oidx] = c;
            p_out[oidx]   = 1.0f / (1.0f + __expf(-c));
        }
    }
}

extern "C" void kernel_launch(void* const* d_in, const int* in_sizes, int n_in,
                              void* d_out, int out_size, void* d_ws, size_t ws_size,
                              hipStream_t stream) {
    const float* token = (const float*)d_in[0];          // (B,S,H)
    const float* entity = (const float*)d_in[1];         // (B,E,H)
    const unsigned char* tmask = (const unsigned char*)d_in[2]; // (B,S) bool
    const float* Wt = (const float*)d_in[3];             // (H,H)
    const float* bt = (const float*)d_in[4];             // (H)
    const float* We = (const float*)d_in[5];             // (H,H)
    const float* be = (const float*)d_in[6];             // (H)
    const float* wb = (const float*)d_in[7];             // (H)

    float* t_ws = (float*)d_ws;                          // (B*S, H) = 12.6 MB
    float* e_ws = t_ws + (size_t)Bq * Sq * Hq;           // (B*E, H) = 0.4 MB

    // t = token @ Wt^T + bt : M=4096, N=768, K=768
    {
        const int M = Bq * Sq, N = Hq, K = Hq;
        const int nTilesN = N / 64;                      // 12
        const int totalTiles = (M / 16) * nTilesN;       // 3072
        const int blocks = (totalTiles + 7) / 8;         // 384
        gemm_bias_wmma_f32<<<blocks, 256, 0, stream>>>(token, Wt, bt, t_ws, M, N, K,
                                                       nTilesN, totalTiles);
    }
    // e = entity @ We^T + be : M=128, N=768, K=768
    {
        const int M = Bq * Eq, N = Hq, K = Hq;
        const int nTilesN = N / 64;                      // 12
        const int totalTiles = (M / 16) * nTilesN;       // 96
        const int blocks = (totalTiles + 7) / 8;         // 12
        gemm_bias_wmma_f32<<<blocks, 256, 0, stream>>>(entity, We, be, e_ws, M, N, K,
                                                       nTilesN, totalTiles);
    }
    // cls + p
    {
        float* cls_out = (float*)d_out;                  // (B,E,S)
        float* p_out   = cls_out + (size_t)Bq * Eq * Sq; // (B,E,S)
        score_kernel<<<Bq * Sq, 256, 0, stream>>>(t_ws, e_ws, tmask, wb, cls_out, p_out);
    }
}